// ForwardXRayVolumeRenderer_16827681866203
// MI455X (gfx1250) — compile-verified
//
#include <hip/hip_runtime.h>

// ForwardXRayVolumeRenderer for MI455X (gfx1250, wave32).
// One thread = (pixel, depth-segment). Ray-box clipped march, the clipped
// range split dynamically over 4 segments (compositing is associative and
// zero-coverage steps are exact no-ops in fp32 since 1.0f+1e-10f==1.0f).
// x-adjacent voxels fetched as one b64 pair (4 gathers/step instead of 8).
// Ray-basis transform dB = (128*R) * c done with V_WMMA_F32_16X16X4_F32.
// Clip bounds use fast v_rcp_f32 (branchless); exact per-step weights make
// the padded bounds safe.

#define IMG 256
#define NPTS 320
#define SEGS 4
#define PIX_PER_BLK 64

typedef __attribute__((ext_vector_type(2))) float v2f;
typedef __attribute__((ext_vector_type(8))) float v8f;
typedef float f2u __attribute__((ext_vector_type(2), aligned(4)));  // 4B-aligned pair

__global__ __launch_bounds__(256) void xray_render_kernel(
    const float* __restrict__ vol,    // 256^3 fp32
    const float* __restrict__ Rm,     // 3x3
    const float* __restrict__ Tv,     // 3
    const float* __restrict__ focal,  // 1
    float* __restrict__ gray_out,     // 65536, layout [w*256 + h]
    unsigned int* __restrict__ maxcell)
{
    const int tx   = threadIdx.x;          // 0..63  pixel within block (w dir)
    const int seg  = threadIdx.y;          // 0..3   depth segment
    const int w    = blockIdx.x * PIX_PER_BLK + tx;
    const int h    = blockIdx.y;
    const int lane = tx & 31;              // 64*ty is 0 mod 32 -> lane == tx&31

    const float f    = focal[0];
    const float invf = 1.0f / f;
    // xs[w] = 255/256 - w/128 ; ys[h] = 255/256 - h/128
    const float gx = (255.0f / 256.0f) - (float)w * (1.0f / 128.0f);
    const float gy = (255.0f / 256.0f) - (float)h * (1.0f / 128.0f);
    const float c0 = gx * invf;            // camera-ray coord 0
    const float c1 = gy * invf;            // camera-ray coord 1  (c2 == 1)

    // Uniform loads of R, T
    const float R00=Rm[0],R01=Rm[1],R02=Rm[2];
    const float R10=Rm[3],R11=Rm[4],R12=Rm[5];
    const float R20=Rm[6],R21=Rm[7],R22=Rm[8];
    const float T0=Tv[0], T1=Tv[1], T2=Tv[2];

    // ---------------- WMMA ray-basis transform --------------------------
    // D[i][n] = sum_k A[i][k] * B[k][n],  A[i][k] = 128*R[i][k] (i,k<3),
    // B[k][n] = c_k of pixel n (c2 = 1, c3 = 0).
    // A (16x4 f32) layout: lanes 0-15: M=lane, v0=K0, v1=K1;
    //                      lanes16-31: M=lane-16, v0=K2, v1=K3.
    const int m = lane & 15;
    const float Rk0 = (m == 0) ? R00 : ((m == 1) ? R10 : R20);
    const float Rk1 = (m == 0) ? R01 : ((m == 1) ? R11 : R21);
    const float Rk2 = (m == 0) ? R02 : ((m == 1) ? R12 : R22);
    v2f A;
    A.x = (m < 3) ? (128.0f * ((lane < 16) ? Rk0 : Rk2)) : 0.0f;
    A.y = (m < 3 && lane < 16) ? (128.0f * Rk1) : 0.0f;

    // B (4x16 f32) layout: lanes 0-15: N=lane, v0=K0, v1=K1;
    //                      lanes16-31: N=lane-16, v0=K2, v1=K3.
    v2f B1;                                   // pixels 0..15 of this wave
    B1.x = (lane < 16) ? c0 : 1.0f;           // K2 row is c2 == 1
    B1.y = (lane < 16) ? c1 : 0.0f;           // K3 row is 0
    const float c0p = __shfl(c0, (lane & 15) + 16, 32);
    const float c1p = __shfl(c1, (lane & 15) + 16, 32);
    v2f B2;                                   // pixels 16..31
    B2.x = (lane < 16) ? c0p : 1.0f;
    B2.y = (lane < 16) ? c1p : 0.0f;

    v8f Z = {};
    v8f D1 = __builtin_amdgcn_wmma_f32_16x16x4_f32(false, A, false, B1,
                                                   (short)0, Z, false, false);
    v8f D2 = __builtin_amdgcn_wmma_f32_16x16x4_f32(false, A, false, B2,
                                                   (short)0, Z, false, false);

    // D layout: VGPR v, lanes 0-15 -> M=v, N=lane. dB_i(pixel n) = D[i][n].
    const float s0 = __shfl(D2[0], lane & 15, 32);
    const float s1 = __shfl(D2[1], lane & 15, 32);
    const float s2 = __shfl(D2[2], lane & 15, 32);
    const float dB0 = (lane < 16) ? D1[0] : s0;
    const float dB1 = (lane < 16) ? D1[1] : s1;
    const float dB2 = (lane < 16) ? D1[2] : s2;

    // idx_i(d) = dA_i + d*dB_i ; dA_i = 127.5 - 128*(R*T)_i (pixel-uniform)
    const float dA0 = 127.5f - 128.0f * (R00*T0 + R01*T1 + R02*T2);
    const float dA1 = 127.5f - 128.0f * (R10*T0 + R11*T1 + R12*T2);
    const float dA2 = 127.5f - 128.0f * (R20*T0 + R21*T1 + R22*T2);

    // ---------------- ray-box clip, branchless via fast v_rcp_f32 -------
    // Exactness is not required: bounds are padded +/-2 steps and per-step
    // weights are the ground truth (zero coverage => exact no-op in fp32).
    // b==0 -> rcp = inf -> bounds fall out correctly through fminf/fmaxf.
    float dlo = 3.0f, dhi = 9.0f;
    {
        const float a[3] = {dA0, dA1, dA2};
        const float b[3] = {dB0, dB1, dB2};
        #pragma unroll
        for (int i = 0; i < 3; ++i) {
            const float rb = __builtin_amdgcn_rcpf(b[i]);
            const float r0 = (-1.0f  - a[i]) * rb;
            const float r1 = (256.0f - a[i]) * rb;
            dlo = fmaxf(dlo, fminf(r0, r1));
            dhi = fminf(dhi, fmaxf(r0, r1));
        }
    }
    int plo = (int)ceilf ((dlo - 3.0f) * (319.0f / 6.0f)) - 2;  // conservative pad
    int phi = (int)floorf((dhi - 3.0f) * (319.0f / 6.0f)) + 2;
    plo = max(plo, 0);
    phi = min(phi, NPTS - 1);
    const int n   = max(phi - plo + 1, 0);
    const int cnt = (n + SEGS - 1) / SEGS;        // dynamic split of active range
    const int myLo = plo + seg * cnt;
    const int myHi = min(myLo + cnt - 1, phi);

    // ---------------- depth-segment march -------------------------------
    float rgbS = 0.0f;   // sum_p dens_p * T_prev * feat_p  (within segment)
    float Tea  = 1.0f;   // prod (1 + eps - dens)  (== 1 - dens in fp32)
    float Top  = 1.0f;   // prod (1 - dens)        (for opacity)

    for (int p = myLo; p <= myHi; ++p) {
        const float d  = fmaf((float)p, 6.0f / 319.0f, 3.0f);
        const float ix = fmaf(d, dB0, dA0);
        const float iy = fmaf(d, dB1, dA1);
        const float iz = fmaf(d, dB2, dA2);

        const float fx0 = floorf(ix), fy0 = floorf(iy), fz0 = floorf(iz);
        const float fx = ix - fx0, fy = iy - fy0, fz = iz - fz0;
        const int i0x = (int)fx0, i0y = (int)fy0, i0z = (int)fz0;

        // per-axis weights with validity folded
        const float wx0 = (1.0f - fx) * (((unsigned)i0x       < 256u) ? 1.0f : 0.0f);
        const float wx1 = fx          * (((unsigned)(i0x + 1) < 256u) ? 1.0f : 0.0f);
        const float wy0 = (1.0f - fy) * (((unsigned)i0y       < 256u) ? 1.0f : 0.0f);
        const float wy1 = fy          * (((unsigned)(i0y + 1) < 256u) ? 1.0f : 0.0f);
        const float wz0 = (1.0f - fz) * (((unsigned)i0z       < 256u) ? 1.0f : 0.0f);
        const float wz1 = fz          * (((unsigned)(i0z + 1) < 256u) ? 1.0f : 0.0f);

        // constant-0.1 density volume: dens = 0.1 * coverage (factorized)
        const float dens = 0.1f * (wx0 + wx1) * (wy0 + wy1) * (wz0 + wz1);

        // x-pair base + in-pair offsets (handles all clamp cases exactly)
        const int bx   = min(max(i0x, 0), 254);
        const int off0 = (i0x >= 255) ? 1 : 0;   // pair lane for corner x0
        const int off1 = (i0x >= 0)   ? 1 : 0;   // pair lane for corner x1
        const int ry0 = (min(max(i0y,     0), 255)) << 8;
        const int ry1 = (min(max(i0y + 1, 0), 255)) << 8;
        const size_t pz0 = (size_t)(min(max(i0z,     0), 255)) << 16;
        const size_t pz1 = (size_t)(min(max(i0z + 1, 0), 255)) << 16;

        const f2u q00 = *(const f2u*)(vol + pz0 + ry0 + bx);   // (z0,y0)
        const f2u q01 = *(const f2u*)(vol + pz0 + ry1 + bx);   // (z0,y1)
        const f2u q10 = *(const f2u*)(vol + pz1 + ry0 + bx);   // (z1,y0)
        const f2u q11 = *(const f2u*)(vol + pz1 + ry1 + bx);   // (z1,y1)

        const float v000 = off0 ? q00.y : q00.x, v001 = off1 ? q00.y : q00.x;
        const float v010 = off0 ? q01.y : q01.x, v011 = off1 ? q01.y : q01.x;
        const float v100 = off0 ? q10.y : q10.x, v101 = off1 ? q10.y : q10.x;
        const float v110 = off0 ? q11.y : q11.x, v111 = off1 ? q11.y : q11.x;

        const float feat =
            wz0 * (wy0 * fmaf(wx0, v000, wx1 * v001) +
                   wy1 * fmaf(wx0, v010, wx1 * v011)) +
            wz1 * (wy0 * fmaf(wx0, v100, wx1 * v101) +
                   wy1 * fmaf(wx0, v110, wx1 * v111));

        rgbS = fmaf(dens * Tea, feat, rgbS);
        Tea *= (1.0f + 1e-10f - dens);
        Top *= (1.0f - dens);
    }

    // ---------------- combine segments (associative compositing) --------
    __shared__ float sR[SEGS][PIX_PER_BLK];
    __shared__ float sT[SEGS][PIX_PER_BLK];
    __shared__ float sO[SEGS][PIX_PER_BLK];
    sR[seg][tx] = rgbS;
    sT[seg][tx] = Tea;
    sO[seg][tx] = Top;
    __syncthreads();

    if (seg == 0) {
        float r = sR[3][tx];
        r = fmaf(sT[2][tx], r, sR[2][tx]);
        r = fmaf(sT[1][tx], r, sR[1][tx]);
        r = fmaf(sT[0][tx], r, sR[0][tx]);
        const float top  = sO[0][tx] * sO[1][tx] * sO[2][tx] * sO[3][tx];
        const float gray = (3.0f * r + (1.0f - top)) * 0.25f;  // mean of (r,r,r,opacity)
        gray_out[w * IMG + h] = gray;                          // screen transpose layout
        atomicMax(maxcell, __float_as_uint(gray));             // gray >= 0: bit-order == float-order
    }
}

__global__ __launch_bounds__(256) void xray_normalize_kernel(
    float* __restrict__ out, const unsigned int* __restrict__ maxcell)
{
    const int i = blockIdx.x * 256 + threadIdx.x;
    const float mx = __uint_as_float(*maxcell);
    out[i] = (out[i] + 1e-8f) / (mx + 1e-8f);
}

extern "C" void kernel_launch(void* const* d_in, const int* in_sizes, int n_in,
                              void* d_out, int out_size, void* d_ws, size_t ws_size,
                              hipStream_t stream) {
    (void)in_sizes; (void)n_in; (void)out_size; (void)ws_size;
    const float* vol = (const float*)d_in[0];   // image3d: 1x1x256x256x256
    const float* Rm  = (const float*)d_in[1];   // 3x3
    const float* Tv  = (const float*)d_in[2];   // 3
    const float* fo  = (const float*)d_in[3];   // 1
    float* out = (float*)d_out;                 // (1,1,256,256) fp32
    unsigned int* cell = (unsigned int*)d_ws;   // 4-byte global max cell

    hipMemsetAsync(cell, 0, sizeof(unsigned int), stream);

    dim3 blk(PIX_PER_BLK, SEGS);                 // 256 threads = 8 wave32
    dim3 grd(IMG / PIX_PER_BLK, IMG);            // 1024 blocks, 8192 waves
    xray_render_kernel<<<grd, blk, 0, stream>>>(vol, Rm, Tv, fo, out, cell);
    xray_normalize_kernel<<<IMG * IMG / 256, 256, 0, stream>>>(out, cell);
}